// PointNetFeaturePropagation_4080218931825
// MI455X (gfx1250) — compile-verified
//
#include <hip/hip_runtime.h>

typedef float v2f __attribute__((ext_vector_type(2)));
typedef float v4f __attribute__((ext_vector_type(4)));
typedef float v8f __attribute__((ext_vector_type(8)));

#define B_   8
#define N_   8192
#define S_   2048
#define C1_  128
#define C2_  256
#define K0_  384          // C1 + C2
#define H0_  256
#define H1_  128
#define MTOT (B_ * N_)    // 65536 rows
#define STATS_BLOCKS 256
#define STATS_ROWS   (MTOT / STATS_BLOCKS)   // 256

#define KNN_SEG 4                   // threads cooperating per query point
#define KNN_PTS 64                  // points per 256-thread block
#define KNN_SLEN (S_ / KNN_SEG)     // 512 candidates per segment

#define CH   32                     // K-chunk staged per LDS buffer
#define TPAD 34                     // LDS tile row stride (conflict-free, 8B-aligned)

// ---------------------------------------------------------------------------
// Kernel 1: 3-nearest-neighbor search. xyz2 (+ squared norms) cached in LDS.
// 4 threads per query point (512 candidates each); per-segment top-3 lists
// merged through LDS in ascending-s order with strict '<' so ties match
// jax.lax.top_k (lowest index wins). Same |a|^2+|b|^2-2ab expansion as ref.
// ---------------------------------------------------------------------------
__global__ __launch_bounds__(256)
void knn3_kernel(const float* __restrict__ xyz1,
                 const float* __restrict__ xyz2,
                 int* __restrict__ idx, float* __restrict__ wts) {
  __shared__ float sx[S_], sy[S_], sz[S_], sn[S_];       // 32 KB
  __shared__ float md[KNN_PTS][KNN_SEG][3];
  __shared__ int   mi[KNN_PTS][KNN_SEG][3];

  const int blocksPerBatch = N_ / KNN_PTS;               // 128
  const int b  = blockIdx.x / blocksPerBatch;
  const int nc = blockIdx.x % blocksPerBatch;

  const float* x2 = xyz2 + (size_t)b * S_ * 3;
  for (int i = threadIdx.x; i < S_; i += 256) {
    float xx = x2[i * 3 + 0], yy = x2[i * 3 + 1], zz = x2[i * 3 + 2];
    sx[i] = xx; sy[i] = yy; sz[i] = zz;
    sn[i] = xx * xx + yy * yy + zz * zz;
  }
  __syncthreads();

  const int pt  = threadIdx.x >> 2;        // 0..63
  const int seg = threadIdx.x & 3;         // 0..3
  const int n   = nc * KNN_PTS + pt;
  const size_t p = (size_t)b * N_ + n;
  const float x = xyz1[p * 3 + 0], y = xyz1[p * 3 + 1], z = xyz1[p * 3 + 2];
  const float n1 = x * x + y * y + z * z;

  float d0 = 1e30f, d1 = 1e30f, d2 = 1e30f;
  int   i0 = 0,     i1 = 0,     i2 = 0;
  const int s0 = seg * KNN_SLEN;
  for (int s = s0; s < s0 + KNN_SLEN; ++s) {
    float d = n1 + sn[s] - 2.0f * (x * sx[s] + y * sy[s] + z * sz[s]);
    if (d < d0)      { d2 = d1; i2 = i1; d1 = d0; i1 = i0; d0 = d; i0 = s; }
    else if (d < d1) { d2 = d1; i2 = i1; d1 = d;  i1 = s; }
    else if (d < d2) { d2 = d;  i2 = s; }
  }
  md[pt][seg][0] = d0; md[pt][seg][1] = d1; md[pt][seg][2] = d2;
  mi[pt][seg][0] = i0; mi[pt][seg][1] = i1; mi[pt][seg][2] = i2;
  __syncthreads();

  if (seg == 0) {
    float e0 = 1e30f, e1 = 1e30f, e2 = 1e30f;
    int   j0 = 0,     j1 = 0,     j2 = 0;
    for (int sg = 0; sg < KNN_SEG; ++sg) {
      for (int k = 0; k < 3; ++k) {
        float d = md[pt][sg][k];
        int   s = mi[pt][sg][k];
        if (d < e0)      { e2 = e1; j2 = j1; e1 = e0; j1 = j0; e0 = d; j0 = s; }
        else if (d < e1) { e2 = e1; j2 = j1; e1 = d;  j1 = s; }
        else if (d < e2) { e2 = d;  j2 = s; }
      }
    }
    float r0 = 1.0f / (e0 + 1e-8f);
    float r1 = 1.0f / (e1 + 1e-8f);
    float r2 = 1.0f / (e2 + 1e-8f);
    float rs = 1.0f / (r0 + r1 + r2);
    idx[p * 3 + 0] = j0; idx[p * 3 + 1] = j1; idx[p * 3 + 2] = j2;
    wts[p * 3 + 0] = r0 * rs; wts[p * 3 + 1] = r1 * rs; wts[p * 3 + 2] = r2 * rs;
  }
}

// ---------------------------------------------------------------------------
// Kernels 2/4: LDS-staged, double-buffered fp32 WMMA GEMM.
//   Y[M,Nout] = act(A[M,K]) @ W[Nout,K]^T + bias
// Block = 8 waves (4M x 2N), block tile 64M x 128N, wave tile 16M x 64N.
// K is processed in 32-wide chunks staged in LDS (row stride TPAD=34 ->
// conflict-free ds_load_b64 fragment reads, rows 8B-aligned). The global
// loads for chunk c+1 are held in registers across the whole compute of
// chunk c (prefetch window = 32 k-columns), then spilled to the alternate
// LDS buffer between barriers.
// FUSE_INTERP: GEMM0's A-tile is built on the fly: k<128 from pts1, k>=128
// by gathering 3 L2-resident pts2 rows with the interpolation weights
// (eliminates the 100 MB Xin buffer and its write+read traffic).
// FUSE_BN: applies previous layer's BatchNorm+ReLU to A fragments (table in LDS).
// ---------------------------------------------------------------------------
template <bool FUSE_INTERP, bool FUSE_BN>
__global__ __launch_bounds__(256)
void gemm_wmma_kernel(const float* __restrict__ A, int K,
                      const float* __restrict__ W,      // [Nout,K]
                      const float* __restrict__ bias,   // [Nout]
                      const float* __restrict__ aScale, // [K] or null
                      const float* __restrict__ aShift, // [K] or null
                      const float* __restrict__ pts1,   // [MTOT,C1] or null
                      const float* __restrict__ pts2,   // [B,S,C2] or null
                      const int*   __restrict__ idx,    // [MTOT,3] or null
                      const float* __restrict__ wts,    // [MTOT,3] or null
                      float* __restrict__ Y, int Nout) {
  __shared__ float sA[2][64 * TPAD];      // 2 x 8704 B
  __shared__ float sW[2][128 * TPAD];     // 2 x 17408 B
  __shared__ float sScale[H0_], sShift[H0_];

  const int tid   = threadIdx.x;
  const int lane  = tid & 31;
  const int wave  = tid >> 5;
  const int wm    = wave & 3;            // 0..3 (M direction)
  const int wn    = wave >> 2;           // 0..1 (N direction)
  const int llow  = lane & 15;
  const int khalf = (lane >> 4) * 2;     // 0 or 2

  if (FUSE_BN && tid < K) { sScale[tid] = aScale[tid]; sShift[tid] = aShift[tid]; }

  // ---- staging descriptors: A = 2 float4 slots, W = 4 float4 slots -------
  int aSt[2], wSt[4], ac4[2];
  size_t ap[2];
  const float* aptr[2];
  int ab[2], ai[2][3];
  float aw[2][3];
#pragma unroll
  for (int e = 0; e < 2; ++e) {
    const int s = tid + e * 256;
    const int r = s >> 3;                // 0..63
    ac4[e] = (s & 7) * 4;                // 0,4,...,28
    aSt[e] = r * TPAD + ac4[e];
    ap[e]  = (size_t)blockIdx.x * 64 + r;
    if (FUSE_INTERP) {
      ab[e] = (int)(ap[e] / N_);
      ai[e][0] = idx[ap[e] * 3 + 0];
      ai[e][1] = idx[ap[e] * 3 + 1];
      ai[e][2] = idx[ap[e] * 3 + 2];
      aw[e][0] = wts[ap[e] * 3 + 0];
      aw[e][1] = wts[ap[e] * 3 + 1];
      aw[e][2] = wts[ap[e] * 3 + 2];
      aptr[e] = nullptr;
    } else {
      aptr[e] = A + ap[e] * (size_t)K + ac4[e];
    }
  }
  const float* wptr[4];
#pragma unroll
  for (int e = 0; e < 4; ++e) {
    const int s = tid + e * 256;
    const int r = s >> 3;                // 0..127
    const int c4 = (s & 7) * 4;
    wSt[e] = r * TPAD + c4;
    wptr[e] = W + ((size_t)blockIdx.y * 128 + r) * K + c4;
  }

  auto loadChunkA = [&](int kb, v4f ra[2]) {
#pragma unroll
    for (int e = 0; e < 2; ++e) {
      if (FUSE_INTERP) {
        if (kb < C1_) {
          ra[e] = *(const v4f*)(pts1 + ap[e] * C1_ + kb + ac4[e]);
        } else {
          const int c2 = kb - C1_ + ac4[e];
          const float* pb = pts2 + (size_t)ab[e] * S_ * C2_ + c2;
          v4f g0 = *(const v4f*)(pb + (size_t)ai[e][0] * C2_);
          v4f g1 = *(const v4f*)(pb + (size_t)ai[e][1] * C2_);
          v4f g2 = *(const v4f*)(pb + (size_t)ai[e][2] * C2_);
          ra[e] = aw[e][0] * g0 + aw[e][1] * g1 + aw[e][2] * g2;
        }
      } else {
        ra[e] = *(const v4f*)(aptr[e] + kb);
      }
    }
  };
  auto loadChunkW = [&](int kb, v4f rw[4]) {
#pragma unroll
    for (int e = 0; e < 4; ++e) rw[e] = *(const v4f*)(wptr[e] + kb);
  };
  auto storeChunk = [&](int buf, const v4f ra[2], const v4f rw[4]) {
#pragma unroll
    for (int e = 0; e < 2; ++e) {               // paired b64 stores (8B-aligned)
      *(v2f*)&sA[buf][aSt[e]]     = (v2f){ra[e].x, ra[e].y};
      *(v2f*)&sA[buf][aSt[e] + 2] = (v2f){ra[e].z, ra[e].w};
    }
#pragma unroll
    for (int e = 0; e < 4; ++e) {
      *(v2f*)&sW[buf][wSt[e]]     = (v2f){rw[e].x, rw[e].y};
      *(v2f*)&sW[buf][wSt[e] + 2] = (v2f){rw[e].z, rw[e].w};
    }
  };

  v8f acc0 = {}, acc1 = {}, acc2 = {}, acc3 = {};
  v4f ra[2], rw[4];

  loadChunkA(0, ra);
  loadChunkW(0, rw);
  storeChunk(0, ra, rw);
  __syncthreads();

  const int aFragOff = (wm * 16 + llow) * TPAD + khalf;
  const int wFragOff = (wn * 64 + llow) * TPAD + khalf;

  int cur = 0;
  for (int kb = 0; kb < K; kb += CH) {
    const bool haveNext = (kb + CH) < K;
    if (haveNext) {                      // prefetch next chunk into registers
      loadChunkA(kb + CH, ra);
      loadChunkW(kb + CH, rw);
    }
    const float* bufA = sA[cur];
    const float* bufW = sW[cur];
#pragma unroll
    for (int kk = 0; kk < CH; kk += 4) {
      v2f a = *(const v2f*)&bufA[aFragOff + kk];
      if (FUSE_BN) {
        const int kc = kb + kk + khalf;
        a.x = fmaxf(0.0f, a.x * sScale[kc]     + sShift[kc]);
        a.y = fmaxf(0.0f, a.y * sScale[kc + 1] + sShift[kc + 1]);
      }
      v2f b0 = *(const v2f*)&bufW[wFragOff + kk];
      v2f b1 = *(const v2f*)&bufW[wFragOff + 16 * TPAD + kk];
      v2f b2 = *(const v2f*)&bufW[wFragOff + 32 * TPAD + kk];
      v2f b3 = *(const v2f*)&bufW[wFragOff + 48 * TPAD + kk];
      acc0 = __builtin_amdgcn_wmma_f32_16x16x4_f32(false, a, false, b0, (short)0, acc0, false, false);
      acc1 = __builtin_amdgcn_wmma_f32_16x16x4_f32(false, a, false, b1, (short)0, acc1, false, false);
      acc2 = __builtin_amdgcn_wmma_f32_16x16x4_f32(false, a, false, b2, (short)0, acc2, false, false);
      acc3 = __builtin_amdgcn_wmma_f32_16x16x4_f32(false, a, false, b3, (short)0, acc3, false, false);
    }
    if (haveNext) {
      __syncthreads();                   // everyone done reading buf[cur^1]
      storeChunk(cur ^ 1, ra, rw);
      __syncthreads();                   // next chunk visible to all waves
      cur ^= 1;
    }
  }

  // C/D layout: VGPR i, lane<16 -> row m0+i, lane>=16 -> row m0+i+8; col = lane&15.
  const int m0 = blockIdx.x * 64 + wm * 16;
  const int n0 = blockIdx.y * 128 + wn * 64;
  const int rbase = m0 + (lane >> 4) * 8;
  v8f accs[4] = {acc0, acc1, acc2, acc3};
#pragma unroll
  for (int t = 0; t < 4; ++t) {
    const int col = n0 + t * 16 + llow;
    const float bv = bias[col];
#pragma unroll
    for (int i = 0; i < 8; ++i) {
      Y[(size_t)(rbase + i) * Nout + col] = accs[t][i] + bv;
    }
  }
}

// ---------------------------------------------------------------------------
// Deterministic per-channel partial sum / sum-of-squares (blockDim.x == C).
// ---------------------------------------------------------------------------
__global__ void stats_kernel(const float* __restrict__ Y, int C,
                             float* __restrict__ psum, float* __restrict__ psq) {
  const int c = threadIdx.x;
  const size_t rowbase = (size_t)blockIdx.x * STATS_ROWS;
  float s = 0.0f, q = 0.0f;
  for (int r = 0; r < STATS_ROWS; ++r) {
    float v = Y[(rowbase + r) * (size_t)C + c];
    s += v;
    q += v * v;
  }
  psum[blockIdx.x * C + c] = s;
  psq [blockIdx.x * C + c] = q;
}

// ---------------------------------------------------------------------------
// Fold partials (fixed order) -> BN scale/shift per channel.
// ---------------------------------------------------------------------------
__global__ void bn_finalize_kernel(const float* __restrict__ psum,
                                   const float* __restrict__ psq,
                                   const float* __restrict__ g,
                                   const float* __restrict__ beta,
                                   int C, float* __restrict__ scale,
                                   float* __restrict__ shift) {
  const int c = threadIdx.x;
  float s = 0.0f, q = 0.0f;
  for (int b = 0; b < STATS_BLOCKS; ++b) { s += psum[b * C + c]; q += psq[b * C + c]; }
  const float inv  = 1.0f / (float)MTOT;
  const float mean = s * inv;
  const float var  = q * inv - mean * mean;   // population var (matches jnp.var)
  const float sc   = g[c] * rsqrtf(var + 1e-5f);
  scale[c] = sc;
  shift[c] = beta[c] - mean * sc;
}

// ---------------------------------------------------------------------------
// Final in-place BN1 + ReLU on d_out (C = 128).
// ---------------------------------------------------------------------------
__global__ void bnrelu_inplace_kernel(float* __restrict__ Y,
                                      const float* __restrict__ scale,
                                      const float* __restrict__ shift,
                                      size_t total) {
  size_t i = (size_t)blockIdx.x * blockDim.x + threadIdx.x;
  const size_t stride = (size_t)gridDim.x * blockDim.x;
  for (; i < total; i += stride) {
    const int c = (int)(i & (H1_ - 1));
    Y[i] = fmaxf(0.0f, Y[i] * scale[c] + shift[c]);
  }
}

// ---------------------------------------------------------------------------
extern "C" void kernel_launch(void* const* d_in, const int* in_sizes, int n_in,
                              void* d_out, int out_size, void* d_ws, size_t ws_size,
                              hipStream_t stream) {
  const float* xyz1  = (const float*)d_in[0];
  const float* xyz2  = (const float*)d_in[1];
  const float* pts1  = (const float*)d_in[2];
  const float* pts2  = (const float*)d_in[3];
  const float* w0    = (const float*)d_in[4];
  const float* b0    = (const float*)d_in[5];
  const float* g0    = (const float*)d_in[6];
  const float* beta0 = (const float*)d_in[7];
  const float* w1    = (const float*)d_in[8];
  const float* b1    = (const float*)d_in[9];
  const float* g1    = (const float*)d_in[10];
  const float* beta1 = (const float*)d_in[11];
  float* out = (float*)d_out;

  // Workspace carve-up (floats).
  float* ws = (float*)d_ws;
  size_t off = 0;
  float* Y0  = ws + off; off += (size_t)MTOT * H0_;        // 16,777,216
  float* wts = ws + off; off += (size_t)MTOT * 3;
  int*   idx = (int*)(ws + off); off += (size_t)MTOT * 3;
  float* ps0 = ws + off; off += (size_t)STATS_BLOCKS * H0_;
  float* pq0 = ws + off; off += (size_t)STATS_BLOCKS * H0_;
  float* sc0 = ws + off; off += H0_;
  float* sh0 = ws + off; off += H0_;
  float* ps1 = ws + off; off += (size_t)STATS_BLOCKS * H1_;
  float* pq1 = ws + off; off += (size_t)STATS_BLOCKS * H1_;
  float* sc1 = ws + off; off += H1_;
  float* sh1 = ws + off; off += H1_;
  (void)ws_size; (void)in_sizes; (void)n_in; (void)out_size;

  // 1) 3-NN (idx + inverse-distance weights); 4 threads per point.
  knn3_kernel<<<B_ * (N_ / KNN_PTS), 256, 0, stream>>>(xyz1, xyz2, idx, wts);

  // 2) GEMM0 with fused gather/interp/concat A-tiles:
  //    Y0 = [pts1 | interp(pts2)] @ w0^T + b0
  dim3 grid0(MTOT / 64, H0_ / 128);
  gemm_wmma_kernel<true, false><<<grid0, 256, 0, stream>>>(
      nullptr, K0_, w0, b0, nullptr, nullptr, pts1, pts2, idx, wts, Y0, H0_);

  // 3) BN0 statistics (deterministic two-pass)
  stats_kernel<<<STATS_BLOCKS, H0_, 0, stream>>>(Y0, H0_, ps0, pq0);
  bn_finalize_kernel<<<1, H0_, 0, stream>>>(ps0, pq0, g0, beta0, H0_, sc0, sh0);

  // 4) GEMM1 with fused BN0+ReLU on A: d_out = relu(bn(Y0)) @ w1^T + b1 (raw Y1)
  dim3 grid1(MTOT / 64, H1_ / 128);
  gemm_wmma_kernel<false, true><<<grid1, 256, 0, stream>>>(
      Y0, H0_, w1, b1, sc0, sh0, nullptr, nullptr, nullptr, nullptr, out, H1_);

  // 5) BN1 statistics over raw Y1
  stats_kernel<<<STATS_BLOCKS, H1_, 0, stream>>>(out, H1_, ps1, pq1);
  bn_finalize_kernel<<<1, H1_, 0, stream>>>(ps1, pq1, g1, beta1, H1_, sc1, sh1);

  // 6) final BN1 + ReLU in place on d_out
  bnrelu_inplace_kernel<<<4096, 256, 0, stream>>>(out, sc1, sh1,
                                                  (size_t)MTOT * H1_);
}